// Transformer_80212809220607
// MI455X (gfx1250) — compile-verified
//
#include <hip/hip_runtime.h>
#include <hip/hip_bf16.h>

// ---------------------------------------------------------------------------
// Deformable-DETR decoder layer for MI455X (gfx1250), fp32 throughout.
// GEMMs on V_WMMA_F32_16X16X4_F32 (wave32, 16x64 tile per wave, 4 accumulator
// chains, branch-free K-loop). The bulk value-projection GEMM additionally
// stages its weight matrix into LDS via GLOBAL_LOAD_ASYNC_TO_LDS_B128,
// shared by 8 waves per workgroup.
// ---------------------------------------------------------------------------

typedef __attribute__((ext_vector_type(2))) float v2f;
typedef __attribute__((ext_vector_type(8))) float v8f;

#define WMMA_F32X4(a, b, c) \
  __builtin_amdgcn_wmma_f32_16x16x4_f32(false, (a), false, (b), (short)0, (c), false, false)

// Problem constants
#define NQ    300
#define BB    8
#define CC    256
#define MH    8
#define CHD   32
#define LL    4
#define KP    4
#define DFFN  1024
#define NQB   (NQ * BB)     // 2400 rows, (q,b) order everywhere
#define PAD   304           // 19 * 16, padded attention extent

// ---------------------------------------------------------------------------
// Generic GEMM: Cout[r, n] = epilogue( sum_k (A[r,k] (+A2[r,k])) * W[n,k] )
// W is [N, Kd] row-major (nn.Linear weight), i.e. out = x @ W.T + bias.
// Requirements (all call sites satisfy): Mr % 16 == 0, N % 64 == 0,
// Kd % 4 == 0, all pointers 8-byte aligned.
// ---------------------------------------------------------------------------
template <bool HAS_A2, bool HAS_RES, bool RELU>
__global__ __launch_bounds__(32) void gemm_wmma(
    const float* __restrict__ A, const float* __restrict__ A2,
    const float* __restrict__ W, const float* __restrict__ bias,
    const float* __restrict__ residual, float* __restrict__ Cout,
    int Mr, int N, int Kd)
{
  const int lane = threadIdx.x;
  const int hf   = lane >> 4;       // 0: lanes 0-15, 1: lanes 16-31
  const int l16  = lane & 15;
  const int row0 = blockIdx.y * 16;
  const int col0 = blockIdx.x * 64;

  const int arow = min(row0 + l16, Mr - 1);
  const float* __restrict__ Arow  = A + (size_t)arow * Kd;
  const float* __restrict__ A2row = HAS_A2 ? (A2 + (size_t)arow * Kd) : nullptr;

  const float* __restrict__ Wc[4];
#pragma unroll
  for (int t = 0; t < 4; ++t)
    Wc[t] = W + (size_t)min(col0 + t * 16 + l16, N - 1) * Kd;

  v8f acc[4] = {};
#pragma unroll 4
  for (int k0 = 0; k0 < Kd; k0 += 4) {
    const int ka = k0 + 2 * hf;          // A 16x4 layout: v0/v1 hold K=ka, ka+1
    v2f a = *(const v2f*)(Arow + ka);
    if constexpr (HAS_A2) {
      const v2f a2 = *(const v2f*)(A2row + ka);
      a += a2;
    }
    const v2f b0 = *(const v2f*)(Wc[0] + ka);
    const v2f b1 = *(const v2f*)(Wc[1] + ka);
    const v2f b2 = *(const v2f*)(Wc[2] + ka);
    const v2f b3 = *(const v2f*)(Wc[3] + ka);
    acc[0] = WMMA_F32X4(a, b0, acc[0]);
    acc[1] = WMMA_F32X4(a, b1, acc[1]);
    acc[2] = WMMA_F32X4(a, b2, acc[2]);
    acc[3] = WMMA_F32X4(a, b3, acc[3]);
  }

#pragma unroll
  for (int t = 0; t < 4; ++t) {
    const int ocol = col0 + t * 16 + l16;
    const float bv = bias[ocol];
#pragma unroll
    for (int r = 0; r < 8; ++r) {
      const int orow = row0 + r + 8 * hf;
      float v = acc[t][r] + bv;
      if constexpr (HAS_RES) v += residual[(size_t)orow * N + ocol];
      if constexpr (RELU)    v = fmaxf(v, 0.f);
      Cout[(size_t)orow * N + ocol] = v;
    }
  }
}

// ---------------------------------------------------------------------------
// Value-projection GEMM with async weight staging (N = Kd = 256 fixed):
//   Cout[r, n] = sum_k (A[r,k] + A2[r,k]) * W[n,k] + bias[n]
// Block = 256 threads (8 waves) covering 32 rows x 256 cols. W is staged
// per K-chunk of 32 into LDS with GLOBAL_LOAD_ASYNC_TO_LDS_B128 (ASYNCcnt),
// padded row stride 36 floats: 16B-aligned b128 destinations, 8B-aligned
// ds_load_b64 compute reads, conflict-free banks (36*l16 mod 64 distinct).
// Requirements: Mr % 32 == 0.
// ---------------------------------------------------------------------------
#define KCH  32
#define LDSW 36
__global__ __launch_bounds__(256) void gemm_val_async(
    const float* __restrict__ A, const float* __restrict__ A2,
    const float* __restrict__ W, const float* __restrict__ bias,
    float* __restrict__ Cout, int Mr)
{
  __shared__ float sW[CC * LDSW];   // 36 KB

  const int tid  = threadIdx.x;
  const int wv   = tid >> 5;
  const int lane = tid & 31;
  const int hf   = lane >> 4;
  const int l16  = lane & 15;
  const int row0 = blockIdx.x * 32 + (wv & 1) * 16;
  const int col0 = (wv >> 1) * 64;

  const int arow = row0 + l16;                       // Mr % 32 == 0 -> in range
  const float* __restrict__ Arow  = A  + (size_t)arow * CC;
  const float* __restrict__ A2row = A2 + (size_t)arow * CC;

  const unsigned ldsBase = (unsigned)(size_t)&sW[0]; // LDS byte offset

  v8f acc[4] = {};
  for (int kc = 0; kc < CC; kc += KCH) {
    // ---- async stage W[:, kc:kc+32] into LDS (2048 b128 chunks, 8/thread)
#pragma unroll
    for (int i = 0; i < 8; ++i) {
      const int idx = tid + i * 256;                 // 0..2047
      const int c   = idx >> 3;                      // weight row (output col)
      const int kk  = (idx & 7) << 2;                // 0,4,...,28
      const float* g = W + (size_t)c * CC + kc + kk;
      const unsigned ldsa = ldsBase + (unsigned)((c * LDSW + kk) << 2);
      asm volatile("global_load_async_to_lds_b128 %0, %1, off"
                   :: "v"(ldsa), "v"(g) : "memory");
    }
    asm volatile("s_wait_asynccnt 0x0" ::: "memory");
    __syncthreads();

    // ---- compute on the staged chunk
#pragma unroll
    for (int k0 = 0; k0 < KCH; k0 += 4) {
      const int ka = k0 + 2 * hf;
      v2f a = *(const v2f*)(Arow + kc + ka);
      const v2f a2 = *(const v2f*)(A2row + kc + ka);
      a += a2;
      const v2f b0 = *(const v2f*)(&sW[(col0 +  0 + l16) * LDSW + ka]);
      const v2f b1 = *(const v2f*)(&sW[(col0 + 16 + l16) * LDSW + ka]);
      const v2f b2 = *(const v2f*)(&sW[(col0 + 32 + l16) * LDSW + ka]);
      const v2f b3 = *(const v2f*)(&sW[(col0 + 48 + l16) * LDSW + ka]);
      acc[0] = WMMA_F32X4(a, b0, acc[0]);
      acc[1] = WMMA_F32X4(a, b1, acc[1]);
      acc[2] = WMMA_F32X4(a, b2, acc[2]);
      acc[3] = WMMA_F32X4(a, b3, acc[3]);
    }
    __syncthreads();   // protect slab before next chunk's fill (WAR)
  }

#pragma unroll
  for (int t = 0; t < 4; ++t) {
    const int ocol = col0 + t * 16 + l16;
    const float bv = bias[ocol];
#pragma unroll
    for (int r = 0; r < 8; ++r) {
      const int orow = row0 + r + 8 * hf;
      Cout[(size_t)orow * CC + ocol] = acc[t][r] + bv;
    }
  }
}

// ---------------------------------------------------------------------------
// Self-attention scores: S[b,m,q,k] = (1/sqrt(32)) * sum_c Q[q,b,m,c]*K[k,b,m,c]
// qk buffer: [2400, 512]; q cols 0..255, k cols 256..511 (col = m*32 + c).
// Output padded [64][304][304]; pad rows/cols hold clamped duplicates that are
// later zeroed (cols, by softmax) or feed only dropped outputs (rows).
// ---------------------------------------------------------------------------
__global__ __launch_bounds__(32) void attn_scores_wmma(
    const float* __restrict__ qk, float* __restrict__ attn)
{
  const int lane = threadIdx.x;
  const int hf   = lane >> 4;
  const int l16  = lane & 15;
  const int bm   = blockIdx.z;
  const int b    = bm >> 3;
  const int m    = bm & 7;
  const int q0   = blockIdx.y * 16;
  const int k0c  = blockIdx.x * 16;

  const int arow = min(q0 + l16, NQ - 1);
  const int bcol = min(k0c + l16, NQ - 1);

  const float* __restrict__ Qr = qk + ((size_t)arow * BB + b) * 512 + m * CHD;
  const float* __restrict__ Kc = qk + ((size_t)bcol * BB + b) * 512 + 256 + m * CHD;

  v8f acc = {};
#pragma unroll
  for (int c0 = 0; c0 < CHD; c0 += 4) {
    const int ca = c0 + 2 * hf;
    const v2f a = *(const v2f*)(Qr + ca);
    const v2f bvec = *(const v2f*)(Kc + ca);
    acc = WMMA_F32X4(a, bvec, acc);
  }

  const float scale = 0.17677669529663687f;  // 1/sqrt(32)
  float* __restrict__ S = attn + (size_t)bm * PAD * PAD;
#pragma unroll
  for (int r = 0; r < 8; ++r)
    S[(size_t)(q0 + r + 8 * hf) * PAD + k0c + l16] = acc[r] * scale;
}

// ---------------------------------------------------------------------------
// Row softmax over k<300; zero the 4 pad columns so padded WMMA K-loop is exact.
// grid = 64*300, block = 128.
// ---------------------------------------------------------------------------
__global__ __launch_bounds__(128) void softmax_row(float* __restrict__ attn)
{
  const int row = blockIdx.x;
  const int bm  = row / NQ;
  const int q   = row - bm * NQ;
  float* __restrict__ S = attn + (size_t)bm * PAD * PAD + (size_t)q * PAD;

  __shared__ float red[128];
  const int tid = threadIdx.x;

  float mx = -1e30f;
  for (int i = tid; i < NQ; i += 128) mx = fmaxf(mx, S[i]);
  red[tid] = mx; __syncthreads();
  for (int s = 64; s > 0; s >>= 1) { if (tid < s) red[tid] = fmaxf(red[tid], red[tid + s]); __syncthreads(); }
  mx = red[0]; __syncthreads();

  float sum = 0.f;
  for (int i = tid; i < NQ; i += 128) { const float e = expf(S[i] - mx); S[i] = e; sum += e; }
  red[tid] = sum; __syncthreads();
  for (int s = 64; s > 0; s >>= 1) { if (tid < s) red[tid] += red[tid + s]; __syncthreads(); }
  const float inv = 1.f / red[0];

  for (int i = tid; i < NQ; i += 128) S[i] *= inv;
  if (tid < PAD - NQ) S[NQ + tid] = 0.f;
}

// ---------------------------------------------------------------------------
// sa[q,b,m,c] = sum_k attn[b,m,q,k] * V[k,b,m,c]   (K padded to 304, pad attn = 0)
// grid = (2 ch-tiles, 19 q-tiles, 64 bm), block = 32.
// ---------------------------------------------------------------------------
__global__ __launch_bounds__(32) void attn_v_wmma(
    const float* __restrict__ attn, const float* __restrict__ vv,
    float* __restrict__ sa)
{
  const int lane = threadIdx.x;
  const int hf   = lane >> 4;
  const int l16  = lane & 15;
  const int bm   = blockIdx.z;
  const int b    = bm >> 3;
  const int m    = bm & 7;
  const int q0   = blockIdx.y * 16;
  const int ch0  = blockIdx.x * 16;

  const float* __restrict__ Srow =
      attn + (size_t)bm * PAD * PAD + (size_t)(q0 + l16) * PAD;
  const float* __restrict__ Vb = vv + (size_t)b * CC + m * CHD + ch0 + l16;

  v8f acc = {};
#pragma unroll 4
  for (int k0 = 0; k0 < PAD; k0 += 4) {
    const int ka = k0 + 2 * hf;
    const v2f a = *(const v2f*)(Srow + ka);
    v2f bvec;
    bvec.x = Vb[(size_t)min(ka,     NQ - 1) * (BB * CC)];
    bvec.y = Vb[(size_t)min(ka + 1, NQ - 1) * (BB * CC)];
    acc = WMMA_F32X4(a, bvec, acc);
  }

#pragma unroll
  for (int r = 0; r < 8; ++r) {
    const int orow = q0 + r + 8 * hf;
    if (orow < NQ)
      sa[((size_t)orow * BB + b) * CC + m * CHD + ch0 + l16] = acc[r];
  }
}

// ---------------------------------------------------------------------------
// LayerNorm over the 256-wide channel dim. grid = 2400, block = 256.
// ---------------------------------------------------------------------------
__global__ __launch_bounds__(256) void layernorm256(
    const float* __restrict__ x, const float* __restrict__ g,
    const float* __restrict__ b, float* __restrict__ out)
{
  const int row = blockIdx.x;
  const int tid = threadIdx.x;
  __shared__ float s[256];

  const float v = x[(size_t)row * CC + tid];
  s[tid] = v; __syncthreads();
  for (int st = 128; st > 0; st >>= 1) { if (tid < st) s[tid] += s[tid + st]; __syncthreads(); }
  const float mean = s[0] * (1.f / 256.f);
  __syncthreads();

  const float d = v - mean;
  s[tid] = d * d; __syncthreads();
  for (int st = 128; st > 0; st >>= 1) { if (tid < st) s[tid] += s[tid + st]; __syncthreads(); }
  const float var = s[0] * (1.f / 256.f);
  const float r = rsqrtf(var + 1e-5f);

  out[(size_t)row * CC + tid] = d * r * g[tid] + b[tid];
}

// ---------------------------------------------------------------------------
// Softmax over groups of 16 (L*K) within each head. aw: [2400, 128] in-place.
// grid = 2400, block = 128.
// ---------------------------------------------------------------------------
__global__ __launch_bounds__(128) void softmax16(float* __restrict__ aw)
{
  const int row = blockIdx.x;
  const int tid = threadIdx.x;
  __shared__ float s[128];

  const float val = aw[(size_t)row * 128 + tid];
  s[tid] = val; __syncthreads();
  const int base = tid & ~15;
  float mx = -1e30f;
#pragma unroll
  for (int j = 0; j < 16; ++j) mx = fmaxf(mx, s[base + j]);
  const float e = expf(val - mx);
  __syncthreads(); s[tid] = e; __syncthreads();
  float sum = 0.f;
#pragma unroll
  for (int j = 0; j < 16; ++j) sum += s[base + j];
  aw[(size_t)row * 128 + tid] = e / sum;
}

// ---------------------------------------------------------------------------
// Deformable multiscale bilinear sampling + weighted sum.
// grid = 2400*8 (one wave per (q,b,m)), block = 32 (lane = channel).
// ---------------------------------------------------------------------------
__global__ __launch_bounds__(32) void deform_sample(
    const float* __restrict__ off, const float* __restrict__ aw,
    const float* __restrict__ refp, const float* __restrict__ val,
    float* __restrict__ out)
{
  const int m    = blockIdx.x & 7;
  const int row  = blockIdx.x >> 3;   // q*8 + b
  const int b    = row & 7;
  const int lane = threadIdx.x;       // channel within head

  const float rx = refp[(size_t)row * 2 + 0];
  const float ry = refp[(size_t)row * 2 + 1];
  const float* __restrict__ offr = off + (size_t)row * CC + m * (LL * KP * 2);
  const float* __restrict__ awr  = aw + (size_t)row * 128 + m * (LL * KP);

  const int    Wl[4]      = {128, 64, 32, 16};
  const int    Hl[4]      = {128, 64, 32, 16};
  const int    HW[4]      = {16384, 4096, 1024, 256};
  const size_t lvlbase[4] = {0, 131072, 163840, 172032};  // rows (8*HW cumulative)

  float acc = 0.f;
#pragma unroll
  for (int l = 0; l < LL; ++l) {
    const int   wl = Wl[l], hl = Hl[l];
    const float wlf = (float)wl, hlf = (float)hl;
    const float* __restrict__ vb =
        val + (lvlbase[l] + (size_t)b * HW[l]) * CC + m * CHD;
#pragma unroll
    for (int k = 0; k < KP; ++k) {
      const float ox = offr[(l * KP + k) * 2 + 0];
      const float oy = offr[(l * KP + k) * 2 + 1];
      const float a  = awr[l * KP + k];
      const float X  = (rx + ox / wlf) * wlf - 0.5f;
      const float Y  = (ry + oy / hlf) * hlf - 0.5f;
      const float xf = floorf(X), yf = floorf(Y);
      const float dx = X - xf, dy = Y - yf;
      const int x0 = (int)xf, y0 = (int)yf;

      float v00 = 0.f, v10 = 0.f, v01 = 0.f, v11 = 0.f;
      if (x0 >= 0 && x0 < wl && y0 >= 0 && y0 < hl)
        v00 = vb[((size_t)y0 * wl + x0) * CC + lane];
      if (x0 + 1 >= 0 && x0 + 1 < wl && y0 >= 0 && y0 < hl)
        v10 = vb[((size_t)y0 * wl + x0 + 1) * CC + lane];
      if (x0 >= 0 && x0 < wl && y0 + 1 >= 0 && y0 + 1 < hl)
        v01 = vb[((size_t)(y0 + 1) * wl + x0) * CC + lane];
      if (x0 + 1 >= 0 && x0 + 1 < wl && y0 + 1 >= 0 && y0 + 1 < hl)
        v11 = vb[((size_t)(y0 + 1) * wl + x0 + 1) * CC + lane];

      const float s = v00 * (1.f - dx) * (1.f - dy) + v10 * dx * (1.f - dy)
                    + v01 * (1.f - dx) * dy + v11 * dx * dy;
      acc += a * s;
    }
  }
  out[(size_t)row * CC + m * CHD + lane] = acc;
}

// ---------------------------------------------------------------------------
// Host-side orchestration
// ---------------------------------------------------------------------------
extern "C" void kernel_launch(void* const* d_in, const int* in_sizes, int n_in,
                              void* d_out, int out_size, void* d_ws, size_t ws_size,
                              hipStream_t stream)
{
  const float* qo    = (const float*)d_in[0];
  const float* qp    = (const float*)d_in[1];
  const float* refp  = (const float*)d_in[2];
  const float* feat[4] = {(const float*)d_in[3], (const float*)d_in[5],
                          (const float*)d_in[7], (const float*)d_in[9]};
  const float* pos[4]  = {(const float*)d_in[4], (const float*)d_in[6],
                          (const float*)d_in[8], (const float*)d_in[10]};
  const float* in_w  = (const float*)d_in[11];
  const float* in_b  = (const float*)d_in[12];
  const float* out_w = (const float*)d_in[13];
  const float* out_b = (const float*)d_in[14];
  const float* n1g   = (const float*)d_in[15];
  const float* n1b   = (const float*)d_in[16];
  const float* n2g   = (const float*)d_in[17];
  const float* n2b   = (const float*)d_in[18];
  const float* n3g   = (const float*)d_in[19];
  const float* n3b   = (const float*)d_in[20];
  const float* val_w = (const float*)d_in[21];
  const float* val_b = (const float*)d_in[22];
  const float* off_w = (const float*)d_in[23];
  const float* off_b = (const float*)d_in[24];
  const float* att_w = (const float*)d_in[25];
  const float* att_b = (const float*)d_in[26];
  const float* dout_w = (const float*)d_in[27];
  const float* dout_b = (const float*)d_in[28];
  const float* f1w   = (const float*)d_in[29];
  const float* f1b   = (const float*)d_in[30];
  const float* f2w   = (const float*)d_in[31];
  const float* f2b   = (const float*)d_in[32];
  float* outp = (float*)d_out;

  // ---- workspace layout (floats). attn and val are time-disjoint -> union.
  float* wsf = (float*)d_ws;
  size_t o = 0;
  float* qk    = wsf + o; o += (size_t)NQB * 512;    // q|k projections
  float* vv    = wsf + o; o += (size_t)NQB * CC;     // v projection
  float* sa    = wsf + o; o += (size_t)NQB * CC;     // self-attn output (pre out_proj)
  float* x1    = wsf + o; o += (size_t)NQB * CC;     // after LN1
  float* tmp   = wsf + o; o += (size_t)NQB * CC;     // pre-LN buffers
  float* offb  = wsf + o; o += (size_t)NQB * CC;     // sampling offsets
  float* awb   = wsf + o; o += (size_t)NQB * 128;    // attention weights
  float* dattn = wsf + o; o += (size_t)NQB * CC;     // deformable attn output
  float* x2    = wsf + o; o += (size_t)NQB * CC;     // after LN2
  float* hbuf  = wsf + o; o += (size_t)NQB * DFFN;   // FFN hidden
  float* big   = wsf + o;                            // union region
  float* attn  = big;                                // 64*304*304 floats (phase A)
  float* val   = big;                                // 174080*256 floats (phase B)

  const dim3 w32(32), b128(128), b256(256);
  const int rtiles = NQB / 16;  // 150

  // ---- 1-2: QKV projections (q/k use qo+qp, v uses qo) ----
  gemm_wmma<true, false, false><<<dim3(512 / 64, rtiles), w32, 0, stream>>>(
      qo, qp, in_w, in_b, nullptr, qk, NQB, 512, CC);
  gemm_wmma<false, false, false><<<dim3(CC / 64, rtiles), w32, 0, stream>>>(
      qo, nullptr, in_w + 512 * CC, in_b + 512, nullptr, vv, NQB, CC, CC);

  // ---- 3-5: batched self-attention ----
  attn_scores_wmma<<<dim3(PAD / 16, PAD / 16, BB * MH), w32, 0, stream>>>(qk, attn);
  softmax_row<<<dim3(BB * MH * NQ), b128, 0, stream>>>(attn);
  attn_v_wmma<<<dim3(CHD / 16, PAD / 16, BB * MH), w32, 0, stream>>>(attn, vv, sa);

  // ---- 6-7: out_proj + residual + LN1 ----
  gemm_wmma<false, true, false><<<dim3(CC / 64, rtiles), w32, 0, stream>>>(
      sa, nullptr, out_w, out_b, qo, tmp, NQB, CC, CC);
  layernorm256<<<dim3(NQB), b256, 0, stream>>>(tmp, n1g, n1b, x1);

  // ---- 8-10: offsets + attention weights + group softmax ----
  gemm_wmma<false, false, false><<<dim3(CC / 64, rtiles), w32, 0, stream>>>(
      x1, nullptr, off_w, off_b, nullptr, offb, NQB, CC, CC);
  gemm_wmma<false, false, false><<<dim3(128 / 64, rtiles), w32, 0, stream>>>(
      x1, nullptr, att_w, att_b, nullptr, awb, NQB, 128, CC);
  softmax16<<<dim3(NQB), b128, 0, stream>>>(awb);

  // ---- 11: per-level value projection (HBM-bound bulk GEMM, async-LDS W) ----
  const int    lvl_rows[4] = {131072, 32768, 8192, 2048};
  const size_t lvl_base[4] = {0, 131072, 163840, 172032};
  for (int l = 0; l < 4; ++l) {
    gemm_val_async<<<dim3(lvl_rows[l] / 32), b256, 0, stream>>>(
        feat[l], pos[l], val_w, val_b, val + lvl_base[l] * CC, lvl_rows[l]);
  }

  // ---- 12: bilinear sampling + weighted aggregation ----
  deform_sample<<<dim3(NQB * MH), w32, 0, stream>>>(offb, awb, refp, val, dattn);

  // ---- 13: dout + residual + LN2 ----
  gemm_wmma<false, true, false><<<dim3(CC / 64, rtiles), w32, 0, stream>>>(
      dattn, nullptr, dout_w, dout_b, x1, tmp, NQB, CC, CC);
  layernorm256<<<dim3(NQB), b256, 0, stream>>>(tmp, n2g, n2b, x2);

  // ---- 14-15: FFN + residual + LN3 -> output ([Nq,B,C] == (q,b) row order) ----
  gemm_wmma<false, false, true><<<dim3(DFFN / 64, rtiles), w32, 0, stream>>>(
      x2, nullptr, f1w, f1b, nullptr, hbuf, NQB, DFFN, CC);
  gemm_wmma<false, true, false><<<dim3(CC / 64, rtiles), w32, 0, stream>>>(
      hbuf, nullptr, f2w, f2b, x2, tmp, NQB, CC, DFFN);
  layernorm256<<<dim3(NQB), b256, 0, stream>>>(tmp, n3g, n3b, outp);

  (void)in_sizes; (void)n_in; (void)out_size; (void)ws_size;
}